// TRMEncoder_63324997812695
// MI455X (gfx1250) — compile-verified
//
#include <hip/hip_runtime.h>
#include <math.h>

// ---------------------------------------------------------------------------
// Problem constants (from reference)
// ---------------------------------------------------------------------------
#define DIM    512
#define VOCAB  17
#define NSEG   64

typedef __attribute__((ext_vector_type(2))) float v2f;
typedef __attribute__((ext_vector_type(8))) float v8f;

#if __has_builtin(__builtin_amdgcn_wmma_f32_16x16x4_f32)
#define HAVE_WMMA_F32X4 1
#else
#define HAVE_WMMA_F32X4 0
#endif

__device__ __forceinline__ float gelu_exact(float x) {
    // erf-based GELU, matching jax.nn.gelu(approximate=False)
    return 0.5f * x * (1.0f + erff(x * 0.70710678118654752440f));
}

// ---------------------------------------------------------------------------
// Tiny GEMM: C[0..VOCAB)[0..DIM) = post( A(VOCABxDIM) @ B(DIMxDIM) )
//   post = gelu(x + bias)  when FUSE_BIAS_GELU, else identity.
// One wave (32 lanes) computes one 16x16 tile with V_WMMA_F32_16X16X4_F32.
// gridDim.x = DIM/16 (N tiles), gridDim.y = 2 (M tiles covering 17 padded rows)
// blockDim.x = 32 (exactly one wave32; EXEC all-ones around WMMA).
// ---------------------------------------------------------------------------
template <bool FUSE_BIAS_GELU>
__global__ __launch_bounds__(32)
void tiny_gemm_wmma_f32(const float* __restrict__ A,
                        const float* __restrict__ B,
                        const float* __restrict__ bias,
                        float* __restrict__ C) {
    const int lane  = threadIdx.x;   // 0..31
    const int ntile = blockIdx.x;    // 0..31
    const int mtile = blockIdx.y;    // 0..1
    const int half  = lane >> 4;     // 0: lanes 0-15, 1: lanes 16-31
    const int l15   = lane & 15;

    // A-fragment row for this lane (16x4 fp32 layout: lanes 0-15 carry M=0..15,
    // half-wave selects K pair: VGPR0 = K {0,2}, VGPR1 = K {1,3}).
    const int  arow   = mtile * 16 + l15;
    const bool avalid = (arow < VOCAB);
    const int  arowc  = avalid ? arow : (VOCAB - 1);   // clamp: branchless guard
    // B/D column for this lane (N striped across lanes within a VGPR).
    const int  ncol   = ntile * 16 + l15;

    v8f acc = {};

#if HAVE_WMMA_F32X4
    for (int k = 0; k < DIM; k += 4) {
        const int kk = k + 2 * half;
        // A 16x4 fragment (always-execute loads, then select-to-zero)
        const float* ap = A + arowc * DIM + kk;
        float a0 = ap[0];
        float a1 = ap[1];
        a0 = avalid ? a0 : 0.0f;
        a1 = avalid ? a1 : 0.0f;
        v2f af = { a0, a1 };
        // B 4x16 fragment: VGPR0 = B[kk][ncol], VGPR1 = B[kk+1][ncol]
        const float* bp = B + kk * DIM + ncol;
        v2f bf = { bp[0], bp[DIM] };
        // D = A*B + C   (8 args: neg_a, A, neg_b, B, c_mod, C, reuse_a, reuse_b)
        acc = __builtin_amdgcn_wmma_f32_16x16x4_f32(
            /*neg_a=*/false, af, /*neg_b=*/false, bf,
            /*c_mod=*/(short)0, acc, /*reuse_a=*/false, /*reuse_b=*/false);
    }
#else
    // Scalar fallback with identical C/D ownership (should not trigger on gfx1250)
    for (int i = 0; i < 8; ++i) {
        int row = mtile * 16 + i + 8 * half;
        if (row < VOCAB) {
            float s = 0.0f;
            for (int k = 0; k < DIM; ++k) s = fmaf(A[row * DIM + k], B[k * DIM + ncol], s);
            acc[i] = s;
        }
    }
#endif

    // Writeback: C/D layout — VGPR i holds M = i (lanes 0-15) / M = i+8 (16-31),
    // N = lane & 15.
#pragma unroll
    for (int i = 0; i < 8; ++i) {
        const int row = mtile * 16 + i + 8 * half;
        if (row < VOCAB) {
            float v = acc[i];
            if (FUSE_BIAS_GELU) v = gelu_exact(v + bias[ncol]);
            C[row * DIM + ncol] = v;
        }
    }
}

// ---------------------------------------------------------------------------
// Fused per-segment histogram + epilogue:
//   out[s, d] = (sum_v hist[s][v] * Mt[v][d]) / len_s + b2[d]
// One block per segment, blockDim == DIM == 512 (thread == output column d).
// ---------------------------------------------------------------------------
__global__ __launch_bounds__(DIM)
void seg_hist_out(const int* __restrict__ tokens,
                  const int* __restrict__ cu,
                  const float* __restrict__ Mt,   // (GELU(E@W1+b1)) @ W2, 17xDIM
                  const float* __restrict__ b2,
                  float* __restrict__ out) {
    __shared__ int hist[VOCAB];

    const int s   = blockIdx.x;
    const int tid = threadIdx.x;

    if (tid < VOCAB) hist[tid] = 0;
    __syncthreads();

    const int start = cu[s];
    const int len   = cu[s + 1] - start;
    for (int i = tid; i < len; i += DIM) {
        atomicAdd(&hist[tokens[start + i]], 1);
    }
    __syncthreads();

    float acc = 0.0f;
#pragma unroll
    for (int v = 0; v < VOCAB; ++v) {
        acc = fmaf((float)hist[v], Mt[v * DIM + tid], acc);
    }
    out[s * DIM + tid] = acc / (float)len + b2[tid];
}

// ---------------------------------------------------------------------------
// kernel_launch
//   d_in: 0=packed_tokens(i32,T) 1=cu_seq_lens(i32,65) 2=embedding(f32,17*512)
//         3=W1(f32,512*512) 4=b1(f32,512) 5=W2(f32,512*512) 6=b2(f32,512)
//   d_out: f32 [64, 512]
//   d_ws:  G (32x512 f32 slot) | Mt (32x512 f32 slot)  -> 128 KiB
// ---------------------------------------------------------------------------
extern "C" void kernel_launch(void* const* d_in, const int* in_sizes, int n_in,
                              void* d_out, int out_size, void* d_ws, size_t ws_size,
                              hipStream_t stream) {
    const int*   tokens = (const int*)d_in[0];
    const int*   cu     = (const int*)d_in[1];
    const float* emb    = (const float*)d_in[2];
    const float* W1     = (const float*)d_in[3];
    const float* b1     = (const float*)d_in[4];
    const float* W2     = (const float*)d_in[5];
    const float* b2     = (const float*)d_in[6];
    float*       out    = (float*)d_out;

    float* G  = (float*)d_ws;        // rows < VOCAB written
    float* Mt = G + 32 * DIM;        // rows < VOCAB written

    dim3 gemm_grid(DIM / 16, 2);     // 32 N-tiles x 2 M-tiles, one wave each

    // Stage 1: G = gelu(emb @ W1 + b1)          (17 x 512, fp32 WMMA)
    tiny_gemm_wmma_f32<true><<<gemm_grid, 32, 0, stream>>>(emb, W1, b1, G);
    // Stage 2: Mt = G @ W2                      (17 x 512, fp32 WMMA)
    tiny_gemm_wmma_f32<false><<<gemm_grid, 32, 0, stream>>>(G, W2, nullptr, Mt);
    // Stage 3: per-segment histogram + out = hist@Mt/len + b2
    seg_hist_out<<<NSEG, DIM, 0, stream>>>(tokens, cu, Mt, b2, out);
}